// FcosDetectionsCodec_6210522710111
// MI455X (gfx1250) — compile-verified
//
#include <hip/hip_runtime.h>
#include <hip/hip_bf16.h>
#include <math.h>

#define NBOX     100
#define NCLS     80
#define TOTPIX   5456      // 64^2 + 32^2 + 16^2 + 8^2 + 4^2
#define WSSTRIDE 640       // per-batch floats in workspace: x1[128] y1[128] x2[128] y2[128] lab[128]

typedef unsigned int v4u __attribute__((ext_vector_type(4)));
typedef int          v8i __attribute__((ext_vector_type(8)));
typedef int          v4i __attribute__((ext_vector_type(4)));

// ---------------------------------------------------------------------------
// Kernel A: per-batch stable descending argsort by area (invalid -> last),
// compacted+transposed box table into workspace, plus valid count.
// One block per batch, O(N^2) rank sort (100x100 compares, trivial).
// ---------------------------------------------------------------------------
__global__ void __launch_bounds__(128)
fcos_sort_kernel(const float* __restrict__ boxes,
                 const int*   __restrict__ labels,
                 const int*   __restrict__ box_valid,
                 float*       __restrict__ ws, int B, int N) {
  const int b = blockIdx.x;
  const int i = threadIdx.x;
  __shared__ float key[128];
  __shared__ int   cnt;
  float x1 = 0.f, y1 = 0.f, x2 = 0.f, y2 = 0.f, k = -INFINITY;
  int vld = 0;
  if (i == 0) cnt = 0;
  if (i < N) {
    const float* bp = boxes + ((size_t)b * N + i) * 4;
    x1 = bp[0]; y1 = bp[1]; x2 = bp[2]; y2 = bp[3];
    vld = box_valid[(size_t)b * N + i];
    const float area = (x2 - x1) * (y2 - y1);
    k = vld ? area : -INFINITY;
    key[i] = k;
  }
  __syncthreads();
  if (i < N) {
    if (vld) atomicAdd(&cnt, 1);
    int rank = 0;
    for (int j = 0; j < N; ++j) {
      const float kj = key[j];
      rank += ((kj > k) || ((kj == k) && (j < i))) ? 1 : 0;  // stable descending
    }
    float* base = ws + (size_t)b * WSSTRIDE;
    base[      rank] = x1;
    base[128 + rank] = y1;
    base[256 + rank] = x2;
    base[384 + rank] = y2;
    ((int*)base)[512 + rank] = labels[(size_t)b * N + i];
  }
  __syncthreads();
  if (i == 0) ((int*)ws)[(size_t)B * WSSTRIDE + b] = cnt;  // all valid ranks < cnt
}

// ---------------------------------------------------------------------------
// Kernel B: one thread per (batch, level-pixel). Box table staged in LDS via
// the Tensor Data Mover; scan sorted valid boxes largest->smallest so last
// (smallest) masked box wins, matching the reference lax.scan overwrites.
// Class channels tracked as a 96-bit register bitmask; every output element
// is written (harness poisons d_out).
// ---------------------------------------------------------------------------
__global__ void __launch_bounds__(256)
fcos_encode_kernel(const float* __restrict__ ws, float* __restrict__ out, int B) {
  const int chunk = blockIdx.x;   // pixel chunk within batch
  const int b     = blockIdx.y;   // batch

  __shared__ float sbuf[WSSTRIDE];
  __shared__ int   s_nv;

  if (threadIdx.x == 0)
    s_nv = ((const int*)ws)[(size_t)B * WSSTRIDE + b];

#if __has_builtin(__builtin_amdgcn_tensor_load_to_lds) && __has_builtin(__builtin_amdgcn_s_wait_tensorcnt)
  {
    // Tensor DMA: 1-D tile of 640 f32 elements, global -> LDS (D# per ISA 08 §8).
    const unsigned long long ga = (unsigned long long)(uintptr_t)(ws + (size_t)b * WSSTRIDE);
    const unsigned lds_off = (unsigned)(uintptr_t)(void*)sbuf;  // low 32b of flat = LDS offset
    v4u g0;
    g0.x = 1u;                                                  // count=1 (valid descriptor)
    g0.y = lds_off;                                             // lds_addr (bytes)
    g0.z = (unsigned)(ga & 0xffffffffu);                        // global_addr[31:0]
    g0.w = (unsigned)((ga >> 32) & 0x1ffffffu) | (2u << 30);    // global_addr[56:32] | type=2
    v8i g1;
    g1[0] = (int)(2u << 16);                                    // data_size=2 -> 4-byte elems
    g1[1] = (int)(((unsigned)WSSTRIDE & 0xffffu) << 16);        // tensor_dim0[15:0] @bits63:48
    g1[2] = (int)(((unsigned)WSSTRIDE >> 16) | (1u << 16));     // tensor_dim0[31:16] | tensor_dim1=1
    g1[3] = (int)((unsigned)WSSTRIDE << 16);                    // tensor_dim1 hi=0 | tile_dim0=640
    g1[4] = 0;                                                  // tile_dim1=0 (1-D), tile_dim2=0
    g1[5] = WSSTRIDE;                                           // tensor_dim0_stride lo32
    g1[6] = 0;
    g1[7] = 0;
    v4i g2 = {0, 0, 0, 0}, g3 = {0, 0, 0, 0};
    v8i g4 = {0, 0, 0, 0, 0, 0, 0, 0};                          // 6-arg toolchain form
    __builtin_amdgcn_tensor_load_to_lds(g0, g1, g2, g3, g4, 0);
    __builtin_amdgcn_s_wait_tensorcnt(0);                       // s_wait_tensorcnt 0
  }
#else
  for (int t = threadIdx.x; t < WSSTRIDE; t += blockDim.x)
    sbuf[t] = ws[(size_t)b * WSSTRIDE + t];
#endif
  __syncthreads();

  const int p = chunk * 256 + (int)threadIdx.x;
  if (p >= TOTPIX) return;

  int li, rel;
  if      (p < 4096) { li = 0; rel = p;        }
  else if (p < 5120) { li = 1; rel = p - 4096; }
  else if (p < 5376) { li = 2; rel = p - 5120; }
  else if (p < 5440) { li = 3; rel = p - 5376; }
  else               { li = 4; rel = p - 5440; }

  const int   lgtab[5]   = {6, 5, 4, 3, 2};
  const int   cumtab[5]  = {0, 4096, 5120, 5376, 5440};
  const float invtab[5]  = {0.125f, 0.0625f, 0.03125f, 0.015625f, 0.0078125f};
  const float rmintab[5] = {0.f, 64.f, 128.f, 256.f, 512.f};
  const float rmaxtab[5] = {64.f, 128.f, 256.f, 512.f, INFINITY};

  const int   lgw   = lgtab[li];
  const int   Wl    = 1 << lgw;
  const int   x     = rel & (Wl - 1);
  const int   y     = rel >> lgw;
  const int   pxl   = 1 << (2 * lgw);         // Hl*Wl (square grids)
  const int   cum   = cumtab[li];
  const float sf    = (float)(8 << li);
  const float inv_s = invtab[li];
  const float rmin  = rmintab[li], rmax = rmaxtab[li];

  const float fx = (float)x, fy = (float)y;
  const float cx  = (fx + 0.5f) * sf;
  const float cyv = (fy + 0.5f) * sf;

  unsigned m0 = 0u, m1 = 0u, m2 = 0u;        // class bitmask (last-writer wins)
  float cen = 0.f, oL = 0.f, oT = 0.f, oR = 0.f, oB = 0.f;

  const int  nv   = s_nv;
  const int* slab = ((const int*)sbuf) + 512;

  for (int n = 0; n < nv; ++n) {             // sorted largest->smallest; all valid
    const float bx1 = sbuf[n], by1 = sbuf[128 + n];
    const float bx2 = sbuf[256 + n], by2 = sbuf[384 + n];
    const float xf = truncf(bx1 * inv_s + 0.5f);   // trunc (not floor) to match jnp.trunc
    const float xl = truncf(bx2 * inv_s - 0.5f);
    const float yf = truncf(by1 * inv_s + 0.5f);
    const float yl = truncf(by2 * inv_s - 0.5f);
    const bool mask = (fy >= yf) && (fy <= yl) && (fx >= xf) && (fx <= xl);
    if (mask) {
      const float l_ = cx - bx1, t_ = cyv - by1, r_ = bx2 - cx, b_ = by2 - cyv;
      const float num = fminf(l_, r_) * fminf(t_, b_);
      const float den = fmaxf(l_, r_) * fmaxf(t_, b_);
      cen = sqrtf(fmaxf(num / fmaxf(den, 1e-12f), 1e-12f));
      const float mv = fmaxf(fmaxf(l_, t_), fmaxf(r_, b_));
      const bool cond = (mv >= rmin) && (mv < rmax);
      const int lab = slab[n];
      const unsigned bit = 1u << (lab & 31);
      const unsigned sel = cond ? bit : 0u;
      if      (lab < 32) m0 = (m0 & ~bit) | sel;
      else if (lab < 64) m1 = (m1 & ~bit) | sel;
      else               m2 = (m2 & ~bit) | sel;
      if (cond) { oL = l_; oT = t_; oR = r_; oB = b_; }
    }
  }

  // ---- write-out (every element, fully coalesced along x) ----
  const size_t CLS_TOT = (size_t)NCLS * B * TOTPIX;
  const size_t CEN_TOT = (size_t)B * TOTPIX;

  float* cp = out + (size_t)NCLS * B * cum + (size_t)b * NCLS * pxl + rel;
  for (int c = 0; c < 32; ++c) cp[(size_t)c        * pxl] = (float)((m0 >> c) & 1u);
  for (int c = 0; c < 32; ++c) cp[(size_t)(c + 32) * pxl] = (float)((m1 >> c) & 1u);
  for (int c = 0; c < 16; ++c) cp[(size_t)(c + 64) * pxl] = (float)((m2 >> c) & 1u);

  out[CLS_TOT + (size_t)B * cum + (size_t)b * pxl + rel] = cen;

  float* bp = out + CLS_TOT + CEN_TOT + (size_t)4 * B * cum + (size_t)b * 4 * pxl + rel;
  bp[0]               = oL;
  bp[(size_t)pxl]     = oT;
  bp[(size_t)2 * pxl] = oR;
  bp[(size_t)3 * pxl] = oB;
}

// ---------------------------------------------------------------------------
extern "C" void kernel_launch(void* const* d_in, const int* in_sizes, int n_in,
                              void* d_out, int out_size, void* d_ws, size_t ws_size,
                              hipStream_t stream) {
  (void)n_in; (void)out_size; (void)ws_size;
  const float* boxes  = (const float*)d_in[0];
  const int*   labels = (const int*)d_in[1];
  const int*   valid  = (const int*)d_in[2];
  const int N = NBOX;
  const int B = in_sizes[1] / N;               // labels is [B, N]
  float* ws = (float*)d_ws;                    // B*640 floats + B ints (~82 KB)

  fcos_sort_kernel<<<dim3((unsigned)B), dim3(128), 0, stream>>>(
      boxes, labels, valid, ws, B, N);

  const int chunks = (TOTPIX + 255) / 256;     // 22 pixel-chunks per batch
  fcos_encode_kernel<<<dim3((unsigned)chunks, (unsigned)B), dim3(256), 0, stream>>>(
      ws, (float*)d_out, B);
}